// Sparsemax_47699906789765
// MI455X (gfx1250) — compile-verified
//
#include <hip/hip_runtime.h>
#include <stdint.h>

// Sparsemax over last axis: x[B,S,N] f32, N = 1024, rows = B*S = 32768.
//
// Design (MI455X / gfx1250, wave32):
//  * one wave per row, 32 f32/lane register-resident; Michelot fixed-point
//    iteration for the simplex threshold tau (exact at convergence, ~4-7
//    passes on Gaussian data); all reductions are shuffle butterflies.
//  * rows are software-pipelined through LDS using CDNA5 async global->LDS
//    copies (GLOBAL_LOAD_ASYNC_TO_LDS_B128, ASYNCcnt): while the wave computes
//    row r, row r+1 streams into the other half of a per-wave LDS double
//    buffer. Async loads retire in order, so s_wait_asynccnt 8 after issuing
//    the next row guarantees the current row is resident.
//  * non-temporal b128 stores for the write-once output stream.

#define NCOLS 1024
#define EPT 32                    // elements per lane (wave32)
#define WAVES_PER_BLOCK 8
#define THREADS (WAVES_PER_BLOCK * 32)
#define ROWS_PER_WAVE 4

typedef float v4f __attribute__((ext_vector_type(4)));

// Issue one 4 KB row as 8 async b128 copies. Per the ISA pseudocode the
// instruction offset is added to BOTH the LDS and global addresses, so one
// (lds_byte, gaddr) pair per lane covers the row with stepped offsets.
__device__ __forceinline__ void issue_async_row(uint32_t lds_byte, uint64_t gaddr) {
    asm volatile(
        "global_load_async_to_lds_b128 %0, %1, off\n\t"
        "global_load_async_to_lds_b128 %0, %1, off offset:512\n\t"
        "global_load_async_to_lds_b128 %0, %1, off offset:1024\n\t"
        "global_load_async_to_lds_b128 %0, %1, off offset:1536\n\t"
        "global_load_async_to_lds_b128 %0, %1, off offset:2048\n\t"
        "global_load_async_to_lds_b128 %0, %1, off offset:2560\n\t"
        "global_load_async_to_lds_b128 %0, %1, off offset:3072\n\t"
        "global_load_async_to_lds_b128 %0, %1, off offset:3584"
        :: "v"(lds_byte), "v"(gaddr) : "memory");
}

__global__ __launch_bounds__(THREADS) void
sparsemax_rows_kernel(const float* __restrict__ x, float* __restrict__ out,
                      int nrows) {
    __shared__ float buf[WAVES_PER_BLOCK * 2 * NCOLS];   // 64 KB: double buffer / wave

    const int wave = threadIdx.x >> 5;
    const int lane = threadIdx.x & 31;
    const long long row0 =
        ((long long)blockIdx.x * WAVES_PER_BLOCK + wave) * ROWS_PER_WAVE;

    float* __restrict__ b0 = &buf[(wave * 2 + 0) * NCOLS];
    float* __restrict__ b1 = &buf[(wave * 2 + 1) * NCOLS];
    // Flat shared-aperture pointer: low 32 bits == LDS byte offset (ISA 10.2).
    const uint32_t lds0 = (uint32_t)(uintptr_t)(void*)b0 + (uint32_t)lane * 16u;
    const uint32_t lds1 = (uint32_t)(uintptr_t)(void*)b1 + (uint32_t)lane * 16u;

    if (row0 < nrows)
        issue_async_row(lds0, (uint64_t)(uintptr_t)(x + row0 * NCOLS) +
                                  (uint64_t)lane * 16u);

    for (int j = 0; j < ROWS_PER_WAVE; ++j) {
        const long long row = row0 + j;
        if (row >= nrows) break;                  // uniform within the wave

        // Prefetch next row into the other buffer, then wait for current row.
        const bool have_next = (j + 1 < ROWS_PER_WAVE) && (row + 1 < nrows);
        if (have_next) {
            issue_async_row(((j + 1) & 1) ? lds1 : lds0,
                            (uint64_t)(uintptr_t)(x + (row + 1) * NCOLS) +
                                (uint64_t)lane * 16u);
            // 8 newer ops may stay outstanding; in-order retire => row j done.
            asm volatile("s_wait_asynccnt 0x8" ::: "memory");
        } else {
            asm volatile("s_wait_asynccnt 0x0" ::: "memory");
        }

        const float* __restrict__ src = (j & 1) ? b1 : b0;

        // ---- LDS -> registers (ds_load_b128, identity layout) ----
        float v[EPT];
#pragma unroll
        for (int i = 0; i < 8; ++i) {
            const v4f t = reinterpret_cast<const v4f*>(src)[i * 32 + lane];
            v[i * 4 + 0] = t.x;
            v[i * 4 + 1] = t.y;
            v[i * 4 + 2] = t.z;
            v[i * 4 + 3] = t.w;
        }

        // ---- Row max (stability shift, matches reference) ----
        float mx = v[0];
#pragma unroll
        for (int e = 1; e < EPT; ++e) mx = fmaxf(mx, v[e]);
#pragma unroll
        for (int off = 16; off > 0; off >>= 1)
            mx = fmaxf(mx, __shfl_xor(mx, off, 32));

        // ---- Shift + full-row sum (seed tau over full active set) ----
        float sum = 0.0f;
#pragma unroll
        for (int e = 0; e < EPT; ++e) {
            v[e] -= mx;
            sum += v[e];
        }
#pragma unroll
        for (int off = 16; off > 0; off >>= 1) sum += __shfl_xor(sum, off, 32);

        float tau = (sum - 1.0f) / (float)NCOLS;
        int c_prev = NCOLS;

        // ---- Michelot: tau <- (sum_{z>tau} z - 1) / #{z>tau}. Active sets
        // are nested (tau monotone non-decreasing), so unchanged count ==
        // unchanged set == exact fixpoint. Break is wave-uniform.
        for (int it = 0; it < 64; ++it) {
            float s = 0.0f, c = 0.0f;
#pragma unroll
            for (int e = 0; e < EPT; ++e) {
                const bool p = v[e] > tau;
                s += p ? v[e] : 0.0f;
                c += p ? 1.0f : 0.0f;
            }
#pragma unroll
            for (int off = 16; off > 0; off >>= 1) {
                s += __shfl_xor(s, off, 32);
                c += __shfl_xor(c, off, 32);
            }
            const int ci = (int)c;             // count <= 1024: exact in f32
            const float tn = (s - 1.0f) / c;   // c >= 1 (z_max = 0 > tau)
            if (ci == c_prev) break;           // stabilized: tau == tn
            c_prev = ci;
            tau = tn;
        }

        // ---- Projection + non-temporal b128 stores (write-once stream) ----
        float* __restrict__ outr = out + row * (long long)NCOLS;
#pragma unroll
        for (int i = 0; i < 8; ++i) {
            v4f t;
            t.x = fmaxf(v[i * 4 + 0] - tau, 0.0f);
            t.y = fmaxf(v[i * 4 + 1] - tau, 0.0f);
            t.z = fmaxf(v[i * 4 + 2] - tau, 0.0f);
            t.w = fmaxf(v[i * 4 + 3] - tau, 0.0f);
            __builtin_nontemporal_store(t, &reinterpret_cast<v4f*>(outr)[i * 32 + lane]);
        }
    }
}

extern "C" void kernel_launch(void* const* d_in, const int* in_sizes, int n_in,
                              void* d_out, int out_size, void* d_ws, size_t ws_size,
                              hipStream_t stream) {
    (void)n_in; (void)out_size; (void)d_ws; (void)ws_size;
    const float* x = (const float*)d_in[0];
    float* out = (float*)d_out;
    const int nrows = in_sizes[0] / NCOLS;  // 32768 for [16,2048,1024]
    const int rows_per_block = WAVES_PER_BLOCK * ROWS_PER_WAVE;
    const int nblocks = (nrows + rows_per_block - 1) / rows_per_block;
    sparsemax_rows_kernel<<<nblocks, THREADS, 0, stream>>>(x, out, nrows);
}